// AOL_v_3676492005801
// MI455X (gfx1250) — compile-verified
//
#include <hip/hip_runtime.h>
#include <math.h>

// Reference shapes
#define NS    128
#define DIM   2048
#define CPD   256
#define HW    128          // H*W = 16*8
#define KNN   5
#define FLATK (DIM * HW)   // 262144

typedef float v2f __attribute__((ext_vector_type(2)));
typedef float v8f __attribute__((ext_vector_type(8)));

// =====================================================================
// Gram matrix  G[i][j] = flat[i] . flat[j]   (flat = inputs, 128 x 262144)
// =====================================================================
#define GK 256
__global__ __launch_bounds__(256) void gram_kernel(const float* __restrict__ flat,
                                                   float* __restrict__ G) {
  __shared__ float As[16][GK + 1];
  __shared__ float Bs[16][GK + 1];
  const int bi = blockIdx.x, bj = blockIdx.y;
  const int tid = threadIdx.x;
  const int ti = tid & 15, tj = tid >> 4;
  float acc = 0.f;
  for (int kc = 0; kc < FLATK; kc += GK) {
#pragma unroll
    for (int it = 0; it < 4; ++it) {
      int lin = tid + it * 256;       // 1024 float4 slots per matrix
      int r   = lin >> 6;             // 64 float4 per row of 256 floats
      int c4  = (lin & 63) << 2;
      float4 a = *(const float4*)(flat + (size_t)(bi * 16 + r) * FLATK + kc + c4);
      As[r][c4 + 0] = a.x; As[r][c4 + 1] = a.y; As[r][c4 + 2] = a.z; As[r][c4 + 3] = a.w;
      float4 b = *(const float4*)(flat + (size_t)(bj * 16 + r) * FLATK + kc + c4);
      Bs[r][c4 + 0] = b.x; Bs[r][c4 + 1] = b.y; Bs[r][c4 + 2] = b.z; Bs[r][c4 + 3] = b.w;
    }
    __syncthreads();
#pragma unroll 16
    for (int k = 0; k < GK; ++k) acc = fmaf(As[ti][k], Bs[tj][k], acc);
    __syncthreads();
  }
  G[(size_t)(bi * 16 + ti) * NS + (bj * 16 + tj)] = acc;
}

// =====================================================================
// kNN + reciprocal-neighbor selection (mirrors the JAX logic exactly)
// =====================================================================
__global__ __launch_bounds__(128) void knn_kernel(const float* __restrict__ G,
                                                  int* __restrict__ neigh) {
  __shared__ float diag[NS];
  __shared__ int fkn_s[NS][KNN];
  const int i = threadIdx.x;
  diag[i] = G[(size_t)i * NS + i];
  __syncthreads();

  float bd[KNN];
  int   bix[KNN];
#pragma unroll
  for (int m = 0; m < KNN; ++m) { bd[m] = 3.4e38f; bix[m] = 0; }
  const float di = diag[i];
  for (int j = 0; j < NS; ++j) {
    float d = di + diag[j] - 2.f * G[(size_t)i * NS + j];
    if (d < bd[KNN - 1]) {           // strict < == top_k tie-break (low idx wins)
      int pos = KNN - 1;
      while (pos > 0 && d < bd[pos - 1]) {
        bd[pos] = bd[pos - 1]; bix[pos] = bix[pos - 1]; --pos;
      }
      bd[pos] = d; bix[pos] = j;
    }
  }
#pragma unroll
  for (int m = 0; m < KNN; ++m) fkn_s[i][m] = bix[m];
  __syncthreads();

  int fi[KNN]; int cnt = 0;
#pragma unroll
  for (int j = 0; j < KNN; ++j) {
    int fj = fkn_s[i][j];
    bool r = false;
#pragma unroll
    for (int kk = 0; kk < KNN; ++kk) r = r || (fkn_s[fj][kk] == i);
    if (r) fi[cnt++] = j;
  }
  if (cnt == 0) { cnt = 1; fi[0] = 0; }   // unreachable: self is always reciprocal
#pragma unroll
  for (int t = 0; t < KNN; ++t) neigh[i * KNN + t] = fkn_s[i][fi[t % cnt]];
}

// =====================================================================
// fp32 WMMA GEMM:   Y[n] (O x 128) = W (O x Kdim) @ B_n (Kdim x 128)
//   MODE 0: B_n = inputs[n]                     (plain store -> feat_cp)
//   MODE 1: B_n = gather(feat_cp, neigh[n])     (relu epilogue -> s_cpct)
//   MODE 2: B_n = s_cpct[n]                     (out = x * (1 + sigmoid))
//
// LDS layout is K-pair interleaved: element (k,p) lives at float index
//   ((k>>1)*PR + p)*2 + (k&1)
// so each lane's B fragment {row k+koff, row k+koff+1} is ONE contiguous
// 8-byte ds_load_b64 straight into the WMMA operand pair (no repacking
// movs).  PR = 144 (== 16 mod 32) puts the two half-waves' b64 loads on
// disjoint 32-bank sets.  Double-buffered chunks overlap global staging
// with the WMMA stream and halve the barrier count.
// =====================================================================
#define KC  32
#define PR  144                 // pair-row stride in column pairs
#define LDSF ((KC / 2) * PR * 2)

template <int MODE>
__global__ __launch_bounds__(256) void gemm_wmma(const float* __restrict__ W,
                                                 const float* __restrict__ X,
                                                 float* __restrict__ Y,
                                                 const float* __restrict__ Xin,
                                                 const int* __restrict__ neigh,
                                                 int O, int Kdim) {
  __shared__ float bs[2][LDSF];   // 2 x 18 KB

  const int n     = blockIdx.y;
  const int obase = blockIdx.x * 128;
  const int tid   = threadIdx.x;
  const int wave  = tid >> 5;
  const int lane  = tid & 31;
  const int wl    = lane & 15;            // A row / B,C column within 16
  const int half  = lane >> 4;            // half-wave -> K offset {0,2}
  const int koff  = half << 1;
  const int arow  = obase + wave * 16 + wl;

  auto stage = [&](int buf, int kc) {
#pragma unroll
    for (int it = 0; it < 4; ++it) {
      int lin = tid + it * 256;          // 1024 float4 slots
      int m   = lin >> 6;                // pair-row 0..15 (K rows 2m, 2m+1)
      int p2  = (lin & 63) << 1;         // even column
      int c0  = kc + (m << 1);
      const float* base;
      if (MODE == 1) {
        // 32-aligned chunk never crosses a 256-channel block -> t uniform
        int t   = c0 >> 8;
        int idx = neigh[n * KNN + t];
        base = X + ((size_t)idx * CPD + (c0 & 255)) * HW;
      } else {
        base = X + ((size_t)n * Kdim + c0) * HW;
      }
      float2 a = *(const float2*)(base + p2);          // row c0
      float2 b = *(const float2*)(base + HW + p2);     // row c0+1
      float4 v = make_float4(a.x, b.x, a.y, b.y);      // K-interleaved
      *(float4*)&bs[buf][(m * PR + p2) * 2] = v;
    }
  };

  v8f acc[8] = {};
  const int nch = Kdim / KC;
  stage(0, 0);
  for (int i = 0; i < nch; ++i) {
    __syncthreads();
    if (i + 1 < nch) stage((i + 1) & 1, (i + 1) * KC);
    const int ib = i & 1;
    const int kc = i * KC;
#pragma unroll
    for (int k = 0; k < KC; k += 4) {
      // A fragment: 16x4 f32, lane wl = row, half-waves hold K {0,1}/{2,3}
      v2f a = *(const v2f*)(W + (size_t)arow * Kdim + kc + k + koff);
      const int kp = (k >> 1) + half;    // pair-row for this half-wave
#pragma unroll
      for (int j = 0; j < 8; ++j) {
        v2f b = *(const v2f*)&bs[ib][kp * (PR * 2) + (j * 16 + wl) * 2];
        acc[j] = __builtin_amdgcn_wmma_f32_16x16x4_f32(
            false, a, false, b, (short)0, acc[j], false, false);
      }
    }
  }

  // ---- epilogue: C/D layout = VGPR r, lanes<16 -> M=r, lanes>=16 -> M=8+r
  const int orow = obase + wave * 16 + (half << 3);
#pragma unroll
  for (int j = 0; j < 8; ++j) {
#pragma unroll
    for (int r = 0; r < 8; ++r) {
      float v = acc[j][r];
      size_t off = ((size_t)n * O + orow + r) * HW + j * 16 + wl;
      if (MODE == 1) v = fmaxf(v, 0.f);
      if (MODE == 2) {
        float s = 1.f / (1.f + __expf(-v));
        v = Xin[off] * (1.f + s);
      }
      Y[off] = v;
    }
  }
}

// =====================================================================
// Host: inputs order = {inputs, labels, w_r, w_list, w_conv}
// workspace: feat_cp 16MB | s_cpct 128MB | G 64KB | neigh
// =====================================================================
extern "C" void kernel_launch(void* const* d_in, const int* in_sizes, int n_in,
                              void* d_out, int out_size, void* d_ws, size_t ws_size,
                              hipStream_t stream) {
  const float* inputs = (const float*)d_in[0];
  /* labels d_in[1] unused in forward */
  const float* w_r    = (const float*)d_in[2];
  const float* w_list = (const float*)d_in[3];
  const float* w_conv = (const float*)d_in[4];
  float* out = (float*)d_out;

  char* ws = (char*)d_ws;
  float* feat_cp = (float*)(ws);                                   // 128*256*128*4  = 16 MB
  float* s_cpct  = (float*)(ws + (size_t)16  * 1024 * 1024);       // 128*2048*128*4 = 128 MB
  float* G       = (float*)(ws + (size_t)144 * 1024 * 1024);       // 128*128*4      = 64 KB
  int*   neigh   = (int*)  (ws + (size_t)144 * 1024 * 1024 + 64 * 1024);

  // 1) Gram matrix + kNN/reciprocal selection
  gram_kernel<<<dim3(8, 8), 256, 0, stream>>>(inputs, G);
  knn_kernel<<<1, 128, 0, stream>>>(G, neigh);

  // 2) feat_cp[n] = w_r @ inputs[n]            (256 x 2048) @ (2048 x 128)
  gemm_wmma<0><<<dim3(2, NS), 256, 0, stream>>>(w_r, inputs, feat_cp,
                                                nullptr, nullptr, CPD, DIM);
  // 3) s_cpct[n] = relu(w_list @ gather(...))  (2048 x 1280) @ (1280 x 128)
  gemm_wmma<1><<<dim3(16, NS), 256, 0, stream>>>(w_list, feat_cp, s_cpct,
                                                 nullptr, neigh, DIM, KNN * CPD);
  // 4) out[n] = inputs[n] * (1 + sigmoid(w_conv @ s_cpct[n]))
  gemm_wmma<2><<<dim3(16, NS), 256, 0, stream>>>(w_conv, s_cpct, out,
                                                 inputs, nullptr, DIM, DIM);
}